// Normal_SelfAttention_89069031784945
// MI455X (gfx1250) — compile-verified
//
#include <hip/hip_runtime.h>
#include <math.h>

// MI455X / gfx1250, wave32. Shapes from the reference:
// B=4, N=2048, D_IN=64, H=8, QKDIM=256 (DQK=32), HIDDEN=512 (DV=64).
// Only softmax row q=0 survives into the output -> prune full NxN attention.

typedef __attribute__((ext_vector_type(2))) float v2f;
typedef __attribute__((ext_vector_type(8))) float v8f;

#define BATCH 4
#define NSEQ  2048
#define BN    (BATCH * NSEQ)   // 8192 rows of x, treated as (8192, 64)
#define DIN   64
#define CK    256              // QKDIM
#define CV    512              // HIDDEN

// ---------------------------------------------------------------------------
// Kernel A: K = x@Wk + bk (8192x256) and V = x@Wv + bv (8192x512) via
// V_WMMA_F32_16X16X4_F32. One wave per 16x16 output tile, 16 accumulating
// WMMAs over the K=64 reduction dim. 4 waves/block, each owning one col tile.
// ---------------------------------------------------------------------------
__global__ void proj_kv_wmma(const float* __restrict__ x,
                             const float* __restrict__ Wk, const float* __restrict__ bk,
                             const float* __restrict__ Wv, const float* __restrict__ bv,
                             float* __restrict__ Kout, float* __restrict__ Vout) {
    const int wave = threadIdx.x >> 5;
    const int lane = threadIdx.x & 31;
    const int lo   = lane & 15;     // N (col) position / M row for A
    const int hi   = lane >> 4;     // half-wave select
    const int rt   = blockIdx.x;                 // row tile: rows 16*rt..16*rt+15
    const int ct   = blockIdx.y * 4 + wave;      // global col tile 0..47

    const float* W; const float* bias; float* out; int C; int c0;
    if (ct < 16) { W = Wk; bias = bk; out = Kout; C = CK; c0 = ct * 16; }
    else         { W = Wv; bias = bv; out = Vout; C = CV; c0 = (ct - 16) * 16; }

    const int row0 = rt * 16;
    const float* arow = x + (size_t)(row0 + lo) * DIN;   // A: row M=lo (M=lo for both halves)

    v8f acc = {};
    #pragma unroll
    for (int kk = 0; kk < 16; ++kk) {
        // A 16x4 f32 layout: VGPR0 = K=(base+2*hi), VGPR1 = K=(base+2*hi+1)
        const int k0 = kk * 4 + 2 * hi;
        v2f a = *(const v2f*)(arow + k0);        // 8B-aligned float2 -> global_load_b64
        // B 4x16 f32 layout: row K striped across lanes within a VGPR
        v2f b;
        b.x = W[(size_t)(k0    ) * C + c0 + lo];
        b.y = W[(size_t)(k0 + 1) * C + c0 + lo];
        // 8 args: (neg_a, A, neg_b, B, c_mod, C, reuse_a, reuse_b)
        acc = __builtin_amdgcn_wmma_f32_16x16x4_f32(false, a, false, b,
                                                    (short)0, acc, false, false);
    }

    const float bval = bias[c0 + lo];
    #pragma unroll
    for (int r = 0; r < 8; ++r) {
        // C/D layout: VGPR r -> M = r + 8*hi, N = lo
        out[(size_t)(row0 + r + 8 * hi) * C + c0 + lo] = acc[r] + bval;
    }
}

// ---------------------------------------------------------------------------
// Kernel B: q0[b, c] = x[b, 0, :] . Wq[:, c] + bq[c]   (only query row 0 is
// ever used by the reference output). 1024 outputs total.
// ---------------------------------------------------------------------------
__global__ void q0_proj(const float* __restrict__ x, const float* __restrict__ Wq,
                        const float* __restrict__ bq, float* __restrict__ q0) {
    const int idx = blockIdx.x * blockDim.x + threadIdx.x;   // 0..1023
    const int b = idx >> 8;
    const int c = idx & 255;
    const float* xr = x + (size_t)b * NSEQ * DIN;            // row n=0 of batch b
    float s = bq[c];
    #pragma unroll
    for (int d = 0; d < DIN; ++d) s += xr[d] * Wq[d * CK + c];
    q0[idx] = s;
}

// ---------------------------------------------------------------------------
// Kernel C: per (b,h): scores = q0 . K^T / sqrt(DV); softmax over 2048 keys;
// ctx = probs . V. One 256-thread block per (b,h); scores live in LDS.
// ---------------------------------------------------------------------------
__global__ void attn_softmax_ctx(const float* __restrict__ q0,
                                 const float* __restrict__ K,
                                 const float* __restrict__ V,
                                 float* __restrict__ out) {
    __shared__ float s_sc[NSEQ];     // 8 KB
    __shared__ float s_red[256];
    __shared__ float s_q[32];

    const int b = blockIdx.x >> 3;
    const int h = blockIdx.x & 7;
    const int t = threadIdx.x;

    if (t < 32) s_q[t] = q0[(b * 8 + h) * 32 + t];
    __syncthreads();

    // scores + local max
    const float scale = 0.125f;      // 1/sqrt(DV) = 1/8
    float lmax = -3.0e38f;
    for (int k = t; k < NSEQ; k += 256) {
        const float* kr = K + ((size_t)(b * NSEQ + k)) * CK + h * 32;
        float s = 0.f;
        #pragma unroll
        for (int d = 0; d < 32; ++d) s += s_q[d] * kr[d];
        s *= scale;
        s_sc[k] = s;
        lmax = fmaxf(lmax, s);
    }
    s_red[t] = lmax; __syncthreads();
    for (int off = 128; off > 0; off >>= 1) {
        if (t < off) s_red[t] = fmaxf(s_red[t], s_red[t + off]);
        __syncthreads();
    }
    const float m = s_red[0];
    __syncthreads();

    // exp + local sum
    float lsum = 0.f;
    for (int k = t; k < NSEQ; k += 256) {
        const float e = __expf(s_sc[k] - m);
        s_sc[k] = e;
        lsum += e;
    }
    s_red[t] = lsum; __syncthreads();
    for (int off = 128; off > 0; off >>= 1) {
        if (t < off) s_red[t] += s_red[t + off];
        __syncthreads();
    }
    const float inv = 1.0f / s_red[0];
    __syncthreads();

    // ctx[b,h,d] = sum_k p[k] * V[b,k,h,d]; 4 threads cooperate per d.
    const int d = t >> 2;
    const int g = t & 3;
    float part = 0.f;
    for (int k = g; k < NSEQ; k += 4) {
        part += s_sc[k] * V[((size_t)(b * NSEQ + k)) * CV + h * 64 + d];
    }
    s_red[t] = part; __syncthreads();
    if (g == 0) {
        out[b * CV + h * 64 + d] =
            (s_red[t] + s_red[t + 1] + s_red[t + 2] + s_red[t + 3]) * inv;
    }
}

// ---------------------------------------------------------------------------
extern "C" void kernel_launch(void* const* d_in, const int* in_sizes, int n_in,
                              void* d_out, int out_size, void* d_ws, size_t ws_size,
                              hipStream_t stream) {
    (void)in_sizes; (void)n_in; (void)out_size; (void)ws_size;
    const float* x  = (const float*)d_in[0];
    const float* Wq = (const float*)d_in[1];
    const float* bq = (const float*)d_in[2];
    const float* Wk = (const float*)d_in[3];
    const float* bk = (const float*)d_in[4];
    const float* Wv = (const float*)d_in[5];
    const float* bv = (const float*)d_in[6];
    float* out = (float*)d_out;

    // Workspace layout (floats): q0 [1024] | K [8192*256] | V [8192*512]
    float* ws = (float*)d_ws;
    float* q0 = ws;
    float* Kb = ws + 1024;
    float* Vb = Kb + (size_t)BN * CK;

    // 512 row tiles x 48 col tiles (16 for K, 32 for V); 4 waves per block.
    proj_kv_wmma<<<dim3(512, 12), 128, 0, stream>>>(x, Wk, bk, Wv, bv, Kb, Vb);
    q0_proj<<<4, 256, 0, stream>>>(x, Wq, bq, q0);
    attn_softmax_ctx<<<32, 256, 0, stream>>>(q0, Kb, Vb, out);
}